// GCN_53197464928416
// MI455X (gfx1250) — compile-verified
//
#include <hip/hip_runtime.h>
#include <stdint.h>

typedef float v2f __attribute__((ext_vector_type(2)));
typedef float v8f __attribute__((ext_vector_type(8)));

#define FIN  128
#define FHID 8
#define FOUT 16

#define TILE_STRIDE 132          // 128 + 4 pad -> A-frag reads hit 16 distinct banks
#define W_STRIDE    130          // 128 + 2 pad -> B-frag b64 reads conflict-free
#define WAVES_PER_BLK 8

// ---------- degree / normalization ----------
__global__ void k_init_deg(float* __restrict__ deg, int N) {
  int i = blockIdx.x * blockDim.x + threadIdx.x;
  if (i < N) deg[i] = 1.0f;  // self-loop contribution
}

__global__ void k_deg_edge(const long long* __restrict__ dst,
                           float* __restrict__ deg, int E) {
  int e = blockIdx.x * blockDim.x + threadIdx.x;
  if (e < E) atomicAdd(&deg[(int)dst[e]], 1.0f);
}

__global__ void k_rsqrt(const float* __restrict__ deg,
                        float* __restrict__ dis, int N) {
  int i = blockIdx.x * blockDim.x + threadIdx.x;
  if (i < N) dis[i] = rsqrtf(deg[i]);  // deg >= 1 always (self-loops)
}

// ---------- GEMM 1: h1[N,8] = x[N,128] @ W1[128,8] ----------
// LDS-staged: W1 transposed+padded (col n -> row of 130 floats, zero for n>=8),
// per-wave 16x128 A tile (coalesced float4 streaming of x),
// then 32 chained V_WMMA_F32_16X16X4_F32 fed purely by ds_load_2addr_b64.
__global__ void __launch_bounds__(256) k_gemm1(const float* __restrict__ x,
                                               const float* __restrict__ W1,
                                               float* __restrict__ h1, int N) {
  __shared__ float sWt[16 * W_STRIDE];                   // W1^T, padded
  __shared__ float sX[WAVES_PER_BLK][16 * TILE_STRIDE];  // per-wave A tiles

  const int tid  = threadIdx.x;
  const int wave = tid >> 5;
  const int lane = tid & 31;

  // stage transposed W1: sWt[n*W_STRIDE + k] = W1[k][n] (zero for n >= 8)
  for (int i = tid; i < 16 * FIN; i += 256) {
    int n = i >> 7, k = i & 127;       // consecutive tid -> consecutive k
    sWt[n * W_STRIDE + k] = (n < FHID) ? W1[k * FHID + n] : 0.0f;
  }

  const int rbase = (blockIdx.x * WAVES_PER_BLK + wave) * 16;
  const bool valid = (rbase < N);

  // coalesced stage of this wave's 16x128 tile: 512 float4s, 32 lanes x 16 iters;
  // each iteration the wave touches 512 contiguous bytes of x.
  if (valid) {
    const float4* __restrict__ xsrc = (const float4*)(x + (size_t)rbase * FIN);
    float* __restrict__ tile = &sX[wave][0];
#pragma unroll
    for (int it = 0; it < 16; ++it) {
      int i = it * 32 + lane;          // all lanes in same row, cols lane*4..+3
      float4 v = xsrc[i];
      int r = i >> 5;
      int cidx = (i & 31) * 4;
      *(float4*)&tile[r * TILE_STRIDE + cidx] = v;
    }
  }
  __syncthreads();

  if (valid) {
    const float* __restrict__ tile = &sX[wave][0];
    const int half = lane >> 4;        // 0: K pair {0,1}; 1: K pair {2,3}
    const int l    = lane & 15;        // A row / BCD col
    const float* __restrict__ arow = &tile[l * TILE_STRIDE + half * 2];
    const float* __restrict__ brow = &sWt[l * W_STRIDE + half * 2];
    v8f c = {};
#pragma unroll
    for (int k = 0; k < FIN; k += 4) {
      v2f a = *(const v2f*)&arow[k];   // ds_load_b64, const imm offset
      v2f b = *(const v2f*)&brow[k];   // ds_load_b64, const imm offset
      c = __builtin_amdgcn_wmma_f32_16x16x4_f32(false, a, false, b,
                                                (short)0, c, false, false);
    }
    if (l < FHID) {
      int row0 = rbase + half * 8;     // D: lanes<16 -> M=0..7, lanes>=16 -> M=8..15
#pragma unroll
      for (int v = 0; v < 8; ++v)
        h1[(size_t)(row0 + v) * FHID + l] = c[v];
    }
  }
}

// ---------- self-loop + bias init for layer-1 aggregation ----------
__global__ void k_init_agg1(const float* __restrict__ h1, const float* __restrict__ dis,
                            const float* __restrict__ b1, float* __restrict__ agg1,
                            int NF) {
  int t = blockIdx.x * blockDim.x + threadIdx.x;
  if (t >= NF) return;
  int i = t >> 3, f = t & 7;
  float di = dis[i];
  agg1[t] = b1[f] + h1[t] * di * di;
}

// ---------- edge scatter-add at F=8 (RELU applied to gathered values if set) ----------
template <int RELU>
__global__ void k_edge_agg(const long long* __restrict__ src,
                           const long long* __restrict__ dst,
                           const float* __restrict__ dis,
                           const float* __restrict__ h,
                           float* __restrict__ out, int E) {
  int e = blockIdx.x * blockDim.x + threadIdx.x;
  if (e >= E) return;
  int s = (int)src[e];
  int d = (int)dst[e];
  float norm = dis[s] * dis[d];
  const float4* __restrict__ hs = (const float4*)(h + (size_t)s * FHID);
  float4 h0 = hs[0];
  float4 h1v = hs[1];
  float v[FHID] = {h0.x, h0.y, h0.z, h0.w, h1v.x, h1v.y, h1v.z, h1v.w};
  float* od = out + (size_t)d * FHID;
#pragma unroll
  for (int f = 0; f < FHID; ++f) {
    float t = v[f];
    if (RELU) t = fmaxf(t, 0.0f);
    atomicAdd(&od[f], t * norm);
  }
}

// ---------- self-loop init for layer-2 aggregation (relu applied here) ----------
__global__ void k_init_agg2(const float* __restrict__ agg1, const float* __restrict__ dis,
                            float* __restrict__ agg2, int NF) {
  int t = blockIdx.x * blockDim.x + threadIdx.x;
  if (t >= NF) return;
  int i = t >> 3;
  float di = dis[i];
  agg2[t] = fmaxf(agg1[t], 0.0f) * di * di;
}

// ---------- GEMM 2: out[N,16] = agg2[N,8] @ W2[8,16] + b2  (f32 WMMA, K=8) ----------
__global__ void k_gemm2(const float* __restrict__ agg2, const float* __restrict__ W2,
                        const float* __restrict__ b2, float* __restrict__ out, int N) {
  int wave = (int)((blockIdx.x * (unsigned)blockDim.x + threadIdx.x) >> 5);
  int lane = threadIdx.x & 31;
  int rbase = wave * 16;
  if (rbase >= N) return;              // wave-uniform: EXEC all-ones inside
  int half = lane >> 4;
  int l    = lane & 15;
  v8f c;
  float bias = b2[l];                  // C[m][n] = b2[n], n = l for both half-waves
#pragma unroll
  for (int v = 0; v < 8; ++v) c[v] = bias;
  const float* arow = agg2 + (size_t)(rbase + l) * FHID;
#pragma unroll
  for (int k = 0; k < FHID; k += 4) {
    int ka = k + half * 2;
    v2f a, b;
    a.x = arow[ka];
    a.y = arow[ka + 1];
    b.x = W2[ka * FOUT + l];
    b.y = W2[(ka + 1) * FOUT + l];
    c = __builtin_amdgcn_wmma_f32_16x16x4_f32(false, a, false, b,
                                              (short)0, c, false, false);
  }
  int row0 = rbase + half * 8;
#pragma unroll
  for (int v = 0; v < 8; ++v)
    out[(size_t)(row0 + v) * FOUT + l] = c[v];
}

extern "C" void kernel_launch(void* const* d_in, const int* in_sizes, int n_in,
                              void* d_out, int out_size, void* d_ws, size_t ws_size,
                              hipStream_t stream) {
  const float*     x    = (const float*)d_in[0];
  const long long* ei   = (const long long*)d_in[1];   // int64 edge_index [2,E]
  const float*     W1   = (const float*)d_in[2];
  const float*     b1   = (const float*)d_in[3];
  const float*     W2   = (const float*)d_in[4];
  const float*     b2   = (const float*)d_in[5];
  float*           outp = (float*)d_out;

  const int N = in_sizes[0] / FIN;
  const int E = in_sizes[1] / 2;
  const long long* srcp = ei;
  const long long* dstp = ei + E;

  // workspace carve-up: deg | dis | h1 | agg1 | agg2  (~20.8 MB, all 16B-aligned)
  char* ws = (char*)d_ws;
  float* deg  = (float*)ws;  ws += (size_t)N * sizeof(float);
  float* dis  = (float*)ws;  ws += (size_t)N * sizeof(float);
  float* h1   = (float*)ws;  ws += (size_t)N * FHID * sizeof(float);
  float* agg1 = (float*)ws;  ws += (size_t)N * FHID * sizeof(float);
  float* agg2 = (float*)ws;

  const int B = 256;
  const int NF = N * FHID;
  const int tiles = (N + 15) / 16;

  k_init_deg <<<(N + B - 1) / B, B, 0, stream>>>(deg, N);
  k_deg_edge <<<(E + B - 1) / B, B, 0, stream>>>(dstp, deg, E);
  k_rsqrt    <<<(N + B - 1) / B, B, 0, stream>>>(deg, dis, N);

  k_gemm1    <<<(tiles + WAVES_PER_BLK - 1) / WAVES_PER_BLK, B, 0, stream>>>(x, W1, h1, N);

  k_init_agg1<<<(NF + B - 1) / B, B, 0, stream>>>(h1, dis, b1, agg1, NF);
  k_edge_agg<0><<<(E + B - 1) / B, B, 0, stream>>>(srcp, dstp, dis, h1, agg1, E);

  k_init_agg2<<<(NF + B - 1) / B, B, 0, stream>>>(agg1, dis, agg2, NF);
  k_edge_agg<1><<<(E + B - 1) / B, B, 0, stream>>>(srcp, dstp, dis, agg1, agg2, E);

  k_gemm2    <<<(tiles * 32 + B - 1) / B, B, 0, stream>>>(agg2, W2, b2, outp, N);
}